// BipartiteGNN_38027640439167
// MI455X (gfx1250) — compile-verified
//
#include <hip/hip_runtime.h>

typedef float v2f __attribute__((ext_vector_type(2)));
typedef float v8f __attribute__((ext_vector_type(8)));

#define HD 64          // hidden / feature dim
#define TILE_M 16      // nodes per WMMA tile
#define LDS_STRIDE 68  // HD + 4 pad -> conflict-free column reads

// ---------------------------------------------------------------- zero fill
__global__ void fill_zero_kernel(float* __restrict__ p, int n) {
  int i = blockIdx.x * blockDim.x + threadIdx.x;
  int stride = gridDim.x * blockDim.x;
  for (; i < n; i += stride) p[i] = 0.0f;
}

// ------------------------------------------------- scatter-add (mean accum)
// 16 threads per edge, each handles a float4 slice of the 64-wide feature row.
__global__ void scatter_accum_kernel(const float* __restrict__ xsrc,
                                     const int* __restrict__ edge,  // [2,E]
                                     float* __restrict__ agg,       // [Ndst,64]
                                     float* __restrict__ cnt,       // [Ndst]
                                     int E) {
  int gid = blockIdx.x * blockDim.x + threadIdx.x;
  int e = gid >> 4;
  if (e >= E) return;
  int part = gid & 15;
  int src = edge[e];
  int dst = edge[E + e];
  const float4 v = *(const float4*)(xsrc + (size_t)src * HD + part * 4);
  float* a = agg + (size_t)dst * HD + part * 4;
  atomicAdd(a + 0, v.x);
  atomicAdd(a + 1, v.y);
  atomicAdd(a + 2, v.z);
  atomicAdd(a + 3, v.w);
  if (part == 0) atomicAdd(cnt + dst, 1.0f);
}

// ---------------------------------------------------- SAGE combine via WMMA
// out = (agg/max(cnt,1)) @ Wl^T + bias + x_dst @ Wr^T, written in place over agg.
// Block = 128 threads (4 waves). Wave w owns output columns [16w, 16w+16).
__global__ void __launch_bounds__(128)
sage_combine_wmma_kernel(float* __restrict__ agg_io,     // [N,64] in: sums, out: result
                         const float* __restrict__ cnt,  // [N]
                         const float* __restrict__ xdst,  // [N,64]
                         const float* __restrict__ Wl,    // [64,64] (row n = out-feats)
                         const float* __restrict__ bias,  // [64]
                         const float* __restrict__ Wr,    // [64,64]
                         int n_nodes) {
  __shared__ float As[TILE_M][LDS_STRIDE];
  __shared__ float Xs[TILE_M][LDS_STRIDE];
  const int tid = threadIdx.x;
  const int m0 = blockIdx.x * TILE_M;

  // Stage tiles: 128 threads x 8 floats = 16x64. All global reads of agg_io
  // happen here, making the in-place overwrite at the end race-free.
  {
    int row = tid >> 3;
    int col = (tid & 7) * 8;
    if (m0 + row < n_nodes) {
      float inv = 1.0f / fmaxf(cnt[m0 + row], 1.0f);
      const float* ga = agg_io + (size_t)(m0 + row) * HD + col;
      const float* gx = xdst + (size_t)(m0 + row) * HD + col;
#pragma unroll
      for (int j = 0; j < 8; ++j) {
        As[row][col + j] = ga[j] * inv;
        Xs[row][col + j] = gx[j];
      }
    } else {
#pragma unroll
      for (int j = 0; j < 8; ++j) {
        As[row][col + j] = 0.0f;
        Xs[row][col + j] = 0.0f;
      }
    }
  }
  __syncthreads();

  const int lane = tid & 31;
  const int half = lane >> 4;  // 0: K pair {k,k+1}, 1: {k+2,k+3}
  const int l16 = lane & 15;
  const int nb = (tid >> 5) * 16;  // column base for this wave

  // D layout: vgpr j holds row (j + 8*half), column (nb + l16). Bias is per column.
  v8f c;
  {
    float bv = bias[nb + l16];
#pragma unroll
    for (int j = 0; j < 8; ++j) c[j] = bv;
  }

#pragma unroll
  for (int kk = 0; kk < HD / 4; ++kk) {
    const int k0 = kk * 4 + 2 * half;
    // A 16x4 fp32: lane<16 -> {K=k,k+1}; lane>=16 -> {K=k+2,k+3}; M = lane%16
    v2f a = {As[l16][k0], As[l16][k0 + 1]};
    // B 4x16 fp32: B[k][n] = Wl[n][k]; N = lane%16 striped across lanes
    v2f bl = {Wl[(nb + l16) * HD + k0], Wl[(nb + l16) * HD + k0 + 1]};
    c = __builtin_amdgcn_wmma_f32_16x16x4_f32(false, a, false, bl, (short)0, c,
                                              false, false);
    v2f x = {Xs[l16][k0], Xs[l16][k0 + 1]};
    v2f br = {Wr[(nb + l16) * HD + k0], Wr[(nb + l16) * HD + k0 + 1]};
    c = __builtin_amdgcn_wmma_f32_16x16x4_f32(false, x, false, br, (short)0, c,
                                              false, false);
  }

  // Store D (in place over agg_io; staging already consumed it).
#pragma unroll
  for (int j = 0; j < 8; ++j) {
    int m = j + 8 * half;
    if (m0 + m < n_nodes) agg_io[(size_t)(m0 + m) * HD + nb + l16] = c[j];
  }
}

// ------------------------------------------------------- column-sum pooling
// Grid stride is a multiple of 64, so each thread always hits one column.
__global__ void col_sum_kernel(const float* __restrict__ x, int n_elems,
                               float* __restrict__ pool) {
  int gid = blockIdx.x * blockDim.x + threadIdx.x;
  int stride = gridDim.x * blockDim.x;
  float s = 0.0f;
  for (int i = gid; i < n_elems; i += stride) s += x[i];
  atomicAdd(pool + (gid & (HD - 1)), s);
}

__global__ void final_head_kernel(const float* __restrict__ pool,
                                  const float* __restrict__ linW,
                                  const float* __restrict__ linb, float inv_total,
                                  float* __restrict__ out) {
  if (threadIdx.x == 0) {
    float s = 0.0f;
    for (int h = 0; h < HD; ++h) s += pool[h] * inv_total * linW[h];
    out[0] = s + linb[0];
  }
}

// ---------------------------------------------------------------- launcher
extern "C" void kernel_launch(void* const* d_in, const int* in_sizes, int n_in,
                              void* d_out, int out_size, void* d_ws, size_t ws_size,
                              hipStream_t stream) {
  const float* x_source = (const float*)d_in[0];
  const float* x_target = (const float*)d_in[1];
  // d_in[2], d_in[3]: edge_attr_* unused by the SAGEConv path
  const float* W_l_s2t = (const float*)d_in[4];
  const float* b_s2t   = (const float*)d_in[5];
  const float* W_r_s2t = (const float*)d_in[6];
  const float* W_l_t2s = (const float*)d_in[7];
  const float* b_t2s   = (const float*)d_in[8];
  const float* W_r_t2s = (const float*)d_in[9];
  const float* lin_W   = (const float*)d_in[10];
  const float* lin_b   = (const float*)d_in[11];
  const int* ei_s2t    = (const int*)d_in[12];
  const int* ei_t2s    = (const int*)d_in[13];

  const int NS = in_sizes[0] / HD;
  const int NT = in_sizes[1] / HD;
  const int E_s2t = in_sizes[12] / 2;
  const int E_t2s = in_sizes[13] / 2;
  const int L = 3;

  // Workspace layout (floats): xs ping-pong, xt ping-pong, counts, pool.
  float* ws = (float*)d_ws;
  const size_t nxs = (size_t)NS * HD;
  const size_t nxt = (size_t)NT * HD;
  float* xs[2] = {ws, ws + nxs};
  float* xt[2] = {ws + 2 * nxs, ws + 2 * nxs + nxt};
  float* cntS = ws + 2 * nxs + 2 * nxt;
  float* cntT = cntS + NS;
  float* pool = cntT + NT;

  const int ZB = 2048, ZT = 256;                      // zero-fill grid
  const int scat_bs = 256;
  const int scat_gs_s2t = (E_s2t * 16 + scat_bs - 1) / scat_bs;
  const int scat_gs_t2s = (E_t2s * 16 + scat_bs - 1) / scat_bs;
  const int comb_gs_t = (NT + TILE_M - 1) / TILE_M;
  const int comb_gs_s = (NS + TILE_M - 1) / TILE_M;

  for (int l = 0; l < L; ++l) {
    const int cur = l & 1, nx = cur ^ 1;
    const float* xs_cur = (l == 0) ? x_source : xs[cur];
    const float* xt_cur = (l == 0) ? x_target : xt[cur];

    fill_zero_kernel<<<ZB, ZT, 0, stream>>>(xt[nx], (int)nxt);
    fill_zero_kernel<<<ZB, ZT, 0, stream>>>(xs[nx], (int)nxs);
    fill_zero_kernel<<<ZB, ZT, 0, stream>>>(cntT, NT);
    fill_zero_kernel<<<ZB, ZT, 0, stream>>>(cntS, NS);

    // aggregate src features into dst accumulators
    scatter_accum_kernel<<<scat_gs_s2t, scat_bs, 0, stream>>>(
        xs_cur, ei_s2t, xt[nx], cntT, E_s2t);
    scatter_accum_kernel<<<scat_gs_t2s, scat_bs, 0, stream>>>(
        xt_cur, ei_t2s, xs[nx], cntS, E_t2s);

    // dense combine (WMMA), in place over the accumulators
    sage_combine_wmma_kernel<<<comb_gs_t, 128, 0, stream>>>(
        xt[nx], cntT, xt_cur, W_l_s2t + (size_t)l * HD * HD, b_s2t + l * HD,
        W_r_s2t + (size_t)l * HD * HD, NT);
    sage_combine_wmma_kernel<<<comb_gs_s, 128, 0, stream>>>(
        xs[nx], cntS, xs_cur, W_l_t2s + (size_t)l * HD * HD, b_t2s + l * HD,
        W_r_t2s + (size_t)l * HD * HD, NS);
  }

  const int fin = ((L - 1) & 1) ^ 1;  // buffer index holding the final layer
  fill_zero_kernel<<<1, 64, 0, stream>>>(pool, HD);
  col_sum_kernel<<<512, 256, 0, stream>>>(xs[fin], (int)nxs, pool);
  col_sum_kernel<<<512, 256, 0, stream>>>(xt[fin], (int)nxt, pool);
  final_head_kernel<<<1, 32, 0, stream>>>(pool, lin_W, lin_b,
                                          1.0f / (float)(NS + NT),
                                          (float*)d_out);
}